// SphericalPooling_34153579937921
// MI455X (gfx1250) — compile-verified
//
#include <hip/hip_runtime.h>
#include <stdint.h>

// Spherical max-pooling, MI455X (gfx1250).
// Phase 1: tiny setup kernel computes per-row (k, pad, t) tables once into d_ws.
// Phase 2: one workgroup per output row; async-DMA (gfx1250 ASYNCcnt path) the
// contiguous 8KB row-pair into LDS with NT hint, build the circular
// sparse-table max up to this row's level t, gather two range-maxes per
// output column, nontemporal store.

#define SP_W       1024
#define SP_H       512
#define SP_H2      256
#define SP_W2      512
#define SP_THREADS 256   // 8 waves (wave32)

typedef float v2f __attribute__((ext_vector_type(2)));

// Per-output-row kernel table, matching the numpy reference in fp64.
__device__ __forceinline__ int3 row_table(int r) {
    const double PI = 3.14159265358979323846264338327950288;
    double theta_sr = (double)SP_H / PI;
    double phi_sr   = (double)SP_W / PI / 2.0;
    double rr  = 1.0 / theta_sr;
    double th  = (2.0 * (double)r + 0.5) / theta_sr;
    double cth = cos(th);
    double sth = sin(th);
    double c   = (cos(rr) - cth * cth) / (sth * sth);
    double width = (c <= -1.0) ? (2.0 * PI) : (2.0 * acos(fmin(c, 1.0)));
    long long kk = (long long)rint(width * phi_sr);   // half-even, like np.round
    if (kk < 2)    kk = 2;
    if (kk > SP_W) kk = SP_W;
    int k = (int)kk;
    return make_int3(k, (k - 1) >> 1, 31 - __clz(k)); // k, pad, floor(log2 k)
}

__global__ void sp_table_kernel(int4* __restrict__ tab) {
    int r = blockIdx.x * 32 + threadIdx.x;
    if (r < SP_H2) {
        int3 q = row_table(r);
        tab[r] = make_int4(q.x, q.y, q.z, 0);
    }
}

__device__ __forceinline__ void async_load_b128_nt(uint32_t lds_off, uint64_t gaddr) {
    // gfx1250 async global->LDS DMA, non-temporal (single-use stream).
    asm volatile("global_load_async_to_lds_b128 %0, %1, off th:TH_LOAD_NT"
                 :
                 : "v"(lds_off), "v"(gaddr)
                 : "memory");
}

template <bool USE_TAB>
__global__ __launch_bounds__(SP_THREADS)
void spherical_pool_kernel(const float* __restrict__ img, float* __restrict__ out,
                           const int4* __restrict__ tab) {
    __shared__ float rows[2 * SP_W];   // staged input row pair
    __shared__ float bufA[SP_W];       // ping
    __shared__ float bufB[SP_W];       // pong
    __shared__ int   s_kpt[3];         // fallback path only

    const int r   = blockIdx.x;   // output row 0..255
    const int bc  = blockIdx.y;   // fused (b, c)
    const int tid = threadIdx.x;  // 0..255

    // ---- kick off async DMA of the contiguous 8KB row pair -----------------
    const float* src = img + ((size_t)bc * SP_H + 2 * (size_t)r) * SP_W;
    {
        uint64_t g = (uint64_t)(uintptr_t)src + (uint64_t)tid * 16u;
        // Low 32 bits of the flat LDS pointer == wave-relative LDS byte offset.
        uint32_t l = (uint32_t)(uintptr_t)(void*)rows + (uint32_t)tid * 16u;
        async_load_b128_nt(l,         g);
        async_load_b128_nt(l + 4096u, g + 4096u);
    }

    int k, pad, t;
    if (USE_TAB) {
        const int4 kt = tab[r];        // uniform -> scalar load path
        k = kt.x; pad = kt.y; t = kt.z;
    } else if (tid == 0) {
        int3 q = row_table(r);
        s_kpt[0] = q.x; s_kpt[1] = q.y; s_kpt[2] = q.z;
    }

    asm volatile("s_wait_asynccnt 0x0" ::: "memory");
    __syncthreads();

    if (!USE_TAB) { k = s_kpt[0]; pad = s_kpt[1]; t = s_kpt[2]; }

    // ---- fused theta-pair reduce + doubling level 1 (t >= 1 always) --------
    // M1[w] = max over rowmax[w], rowmax[w+1] (circular)
#pragma unroll
    for (int i = 0; i < SP_W / SP_THREADS; ++i) {
        int w  = tid + i * SP_THREADS;
        int w1 = (w + 1) & (SP_W - 1);
        float a = fmaxf(rows[w],  rows[w  + SP_W]);
        float b = fmaxf(rows[w1], rows[w1 + SP_W]);
        bufA[w] = fmaxf(a, b);
    }
    __syncthreads();

    // ---- remaining doubling levels 2..t (t uniform per block) --------------
    float* cur = bufA;
    float* nxt = bufB;
    for (int m = 2; m <= t; ++m) {
        const int d = 1 << (m - 1);
#pragma unroll
        for (int i = 0; i < SP_W / SP_THREADS; ++i) {
            int w = tid + i * SP_THREADS;
            nxt[w] = fmaxf(cur[w], cur[(w + d) & (SP_W - 1)]);
        }
        __syncthreads();
        float* tmp = cur; cur = nxt; nxt = tmp;
    }

    // ---- gather: window max of width k = max(M_t[s1], M_t[s1+k-2^t]) -------
    const int po2 = 1 << t;
    const int j0  = 2 * tid;
    int s1a = (2 * j0 - pad) & (SP_W - 1);     // two's complement AND == mod W
    int s2a = (s1a + k - po2) & (SP_W - 1);
    int s1b = (s1a + 2) & (SP_W - 1);
    int s2b = (s1b + k - po2) & (SP_W - 1);

    v2f o;
    o.x = fmaxf(cur[s1a], cur[s2a]);
    o.y = fmaxf(cur[s1b], cur[s2b]);

    v2f* op = (v2f*)(out + ((size_t)bc * SP_H2 + (size_t)r) * SP_W2) + tid;
    __builtin_nontemporal_store(o, op);        // single-use output stream
}

extern "C" void kernel_launch(void* const* d_in, const int* in_sizes, int n_in,
                              void* d_out, int out_size, void* d_ws, size_t ws_size,
                              hipStream_t stream) {
    const float* img = (const float*)d_in[0];
    float*       out = (float*)d_out;

    const int total = in_sizes[0];              // B*C*H*W
    const int bc    = total / (SP_H * SP_W);    // B*C (64 for the reference)

    dim3 grid(SP_H2, bc);
    dim3 block(SP_THREADS);

    if (ws_size >= SP_H2 * sizeof(int4)) {
        int4* tab = (int4*)d_ws;
        sp_table_kernel<<<dim3((SP_H2 + 31) / 32), dim3(32), 0, stream>>>(tab);
        spherical_pool_kernel<true><<<grid, block, 0, stream>>>(img, out, tab);
    } else {
        // Scratch too small: self-contained variant (table inlined per block).
        spherical_pool_kernel<false><<<grid, block, 0, stream>>>(img, out, nullptr);
    }
}